// StatefulMetrics_37031208026335
// MI455X (gfx1250) — compile-verified
//
#include <hip/hip_runtime.h>
#include <cfloat>

typedef float v2f __attribute__((ext_vector_type(2)));
typedef float v4f __attribute__((ext_vector_type(4)));
typedef float v8f __attribute__((ext_vector_type(8)));

// ---------------------------------------------------------------------------
// Kernel 1: streaming argmax-equality count.
// Thread-per-row, 16x b128 non-temporal loads per row (row = 64 f32 = 256 B,
// base is 256B aligned), register argmax with first-occurrence tie-break,
// wave32 shuffle reduction, one atomicAdd per wave.
// ---------------------------------------------------------------------------
__global__ void __launch_bounds__(256)
argmax_count_kernel(const float* __restrict__ y_pred,
                    const int*   __restrict__ y_true,
                    unsigned int* __restrict__ d_count,
                    int n_rows)
{
    int tid    = blockIdx.x * blockDim.x + threadIdx.x;
    int stride = gridDim.x * blockDim.x;

    unsigned int local = 0;
    for (int r = tid; r < n_rows; r += stride) {
        const v4f* row = reinterpret_cast<const v4f*>(y_pred + (size_t)r * 64);
        float best = -FLT_MAX;
        int   bidx = 0;
#pragma unroll
        for (int c = 0; c < 16; ++c) {
            // streamed once; NT hint keeps the 512MB stream out of L2 residency
            v4f v = __builtin_nontemporal_load(row + c);
#pragma unroll
            for (int j = 0; j < 4; ++j) {
                float f = v[j];
                if (f > best) { best = f; bidx = 4 * c + j; }  // first-max wins
            }
        }
        local += (bidx == y_true[r]) ? 1u : 0u;
    }

    // wave32 reduction (gfx1250 is wave32-only; width must be 32, not 64)
#pragma unroll
    for (int off = 16; off > 0; off >>= 1)
        local += __shfl_down(local, off, 32);

    if ((threadIdx.x & 31) == 0 && local != 0)
        atomicAdd(d_count, local);
}

// ---------------------------------------------------------------------------
// Kernel 2: single-wave finalize.
// sum(cm) via V_WMMA_F32_16X16X4_F32 with an all-ones A matrix:
//   D[i][j] = sum_k B[k][j]  =>  sum(D) = 16 * sum(B-tile)
// (total is invariant to the exact B lane/VGPR mapping, so this is exact and
// layout-robust). trace(cm) via direct diagonal gather. One wave => EXEC is
// all ones around the WMMA, as the ISA requires.
// ---------------------------------------------------------------------------
__global__ void __launch_bounds__(32)
finalize_kernel(const float* __restrict__ cm,
                const unsigned int* __restrict__ d_count,
                float* __restrict__ out,
                int n_rows)
{
    const int lane = threadIdx.x;  // 0..31, one wave

    v2f ones; ones[0] = 1.0f; ones[1] = 1.0f;
    v8f acc = {};

    const v2f* cmv = reinterpret_cast<const v2f*>(cm);  // 4096 f32 = 64 tiles of 4x16
#pragma unroll 4
    for (int t = 0; t < 64; ++t) {
        v2f b = cmv[t * 32 + lane];
        // 8 args: (neg_a, A, neg_b, B, c_mod, C, reuse_a, reuse_b)
        acc = __builtin_amdgcn_wmma_f32_16x16x4_f32(
            false, ones, false, b, (short)0, acc, false, false);
    }

    float lane_sum = 0.0f;
#pragma unroll
    for (int i = 0; i < 8; ++i) lane_sum += acc[i];

    // diagonal elements 2*lane and 2*lane+1:  cm[i*64 + i] = cm[i*65]
    float diag = cm[130 * lane] + cm[130 * lane + 65];

#pragma unroll
    for (int off = 16; off > 0; off >>= 1) {
        lane_sum += __shfl_down(lane_sum, off, 32);
        diag     += __shfl_down(diag, off, 32);
    }

    if (lane == 0) {
        float sum_cm = lane_sum * (1.0f / 16.0f);   // undo the 16x replication
        float cnt    = (float)(*d_count);           // < 2^24, exact in f32
        out[0] = (diag + cnt) / (sum_cm + (float)n_rows);
    }
}

// ---------------------------------------------------------------------------
extern "C" void kernel_launch(void* const* d_in, const int* in_sizes, int n_in,
                              void* d_out, int out_size, void* d_ws, size_t ws_size,
                              hipStream_t stream)
{
    const float* y_pred = (const float*)d_in[0];   // [N, 64] f32
    const float* cm     = (const float*)d_in[1];   // [64, 64] f32
    const int*   y_true = (const int*)d_in[2];     // [N] int
    int n_rows = in_sizes[2];

    unsigned int* d_count = (unsigned int*)d_ws;
    hipMemsetAsync(d_ws, 0, sizeof(unsigned int), stream);  // graph-capture safe

    const int threads = 256;                        // 8 waves per block (wave32)
    int blocks = (n_rows + threads - 1) / threads;
    if (blocks > 4096) blocks = 4096;               // grid-stride covers the rest
    if (blocks < 1) blocks = 1;

    argmax_count_kernel<<<blocks, threads, 0, stream>>>(y_pred, y_true, d_count, n_rows);
    finalize_kernel<<<1, 32, 0, stream>>>(cm, d_count, (float*)d_out, n_rows);
}